// PoseDetector_23751169147305
// MI455X (gfx1250) — compile-verified
//
#include <hip/hip_runtime.h>
#include <math.h>
#include <stdint.h>

// ---------------- problem constants ----------------
#define BB 8
#define SS 25
#define HH 512
#define WW 512
#define AA 20
#define KK 100
#define NPTS 10
#define NMAPS (BB * SS)          // 200
#define HWSZ (HH * WW)           // 262144
#define CAP 1024                 // per-map candidate capacity
#define THR_ABS (2.0f / (float)(HH * WW))
#define THR_RELF 0.01f
#define KP_MAX_DISTF 20.0f

// ---------------- output offsets (floats) ----------------
#define OFF_SCORES  ((size_t)0)                       // B*S*K          = 20000
#define OFF_COORDS  ((size_t)20000)                   // B*S*K*2        = 40000
#define OFF_VALID   ((size_t)60000)                   // B*S*K          = 20000
#define OFF_EPAF    ((size_t)80000)                   // B*A*K*K        = 1600000
#define OFF_ER      ((size_t)1680000)
#define OFF_PV      ((size_t)3280000)
#define OFF_IDXPAIR ((size_t)4880000)                 // B*A*2*K*K      = 3200000

typedef __attribute__((ext_vector_type(2))) float v2f;
typedef __attribute__((ext_vector_type(8))) float v8f;
typedef int v4i_vs __attribute__((vector_size(4 * sizeof(int))));

// ---------------- CDNA5 async global->LDS helpers ----------------
#if defined(__HIP_DEVICE_COMPILE__) && \
    __has_builtin(__builtin_amdgcn_global_load_async_to_lds_b128)
#define USE_ASYNC_LDS 1
#endif

__device__ __forceinline__ void async_g2l_b128(const float4* g, float4* l) {
#if defined(USE_ASYNC_LDS)
  // Exact parameter types per compiler diagnostic:
  //   arg0: vector_size(16) int * in AS1 (global), non-const
  //   arg1: vector_size(16) int * in AS3 (LDS)
  // inttoptr round trips: AS1 keeps full 64-bit VA; AS3 keeps low 32 bits
  // (generic shared address carries the LDS offset in addr[31:0]).
  typedef __attribute__((address_space(1))) v4i_vs* as1_v4i;
  typedef __attribute__((address_space(3))) v4i_vs* as3_v4i;
  __builtin_amdgcn_global_load_async_to_lds_b128(
      (as1_v4i)(uintptr_t)g, (as3_v4i)(uintptr_t)l, 0, 0);
#else
  (void)g; (void)l;
#endif
}

__device__ __forceinline__ void wait_asynccnt_le1() {
#if defined(__HIP_DEVICE_COMPILE__)
#if __has_builtin(__builtin_amdgcn_s_wait_asynccnt)
  __builtin_amdgcn_s_wait_asynccnt(1);
#else
  asm volatile("s_wait_asynccnt 0x1" ::: "memory");
#endif
#endif
}

__device__ __forceinline__ void wait_asynccnt_le0() {
#if defined(__HIP_DEVICE_COMPILE__)
#if __has_builtin(__builtin_amdgcn_s_wait_asynccnt)
  __builtin_amdgcn_s_wait_asynccnt(0);
#else
  asm volatile("s_wait_asynccnt 0x0" ::: "memory");
#endif
#endif
}

// =======================================================================
// Kernel 1: per-map softmax statistics (max M, sumexp Z). One WG per map.
// Streams the 1MB map through LDS using ASYNCcnt-tracked async copies,
// double-buffered. Each thread consumes only its own staged 16B slot, so
// no workgroup barriers are needed in the streaming loop: the per-wave
// s_wait_asynccnt orders the async-engine LDS write against the ds_load.
// =======================================================================
__global__ void k_softmax_stats(const float* __restrict__ pose,
                                float* __restrict__ Mout,
                                float* __restrict__ Zout) {
  const int m = blockIdx.x;
  const int tid = threadIdx.x;
  const float4* __restrict__ p4 = (const float4*)(pose + (size_t)m * HWSZ);
  const int n4 = HWSZ / 4;              // 65536 float4
  const int nChunks = n4 / 256;         // 256 chunks x 256 float4
  __shared__ float red[256];

  float mx = -INFINITY;
#if defined(USE_ASYNC_LDS)
  __shared__ float4 stage[2][256];
  // ---- pass 1: max, async double-buffered ----
  async_g2l_b128(p4 + tid, &stage[0][tid]);
  for (int ch = 0; ch < nChunks; ++ch) {
    const int cur = ch & 1;
    if (ch + 1 < nChunks) {
      async_g2l_b128(p4 + (ch + 1) * 256 + tid, &stage[cur ^ 1][tid]);
      wait_asynccnt_le1();
    } else {
      wait_asynccnt_le0();
    }
    float4 v = stage[cur][tid];
    mx = fmaxf(fmaxf(mx, v.x), fmaxf(v.y, fmaxf(v.z, v.w)));
  }
#else
  for (int i = tid; i < n4; i += 256) {
    float4 v = p4[i];
    mx = fmaxf(fmaxf(mx, v.x), fmaxf(v.y, fmaxf(v.z, v.w)));
  }
#endif
  red[tid] = mx;
  __syncthreads();
  for (int s = 128; s > 0; s >>= 1) {
    if (tid < s) red[tid] = fmaxf(red[tid], red[tid + s]);
    __syncthreads();
  }
  const float Mv = red[0];
  __syncthreads();

  float sum = 0.f;
#if defined(USE_ASYNC_LDS)
  // ---- pass 2: sumexp, async double-buffered (data is L2-hot) ----
  async_g2l_b128(p4 + tid, &stage[0][tid]);
  for (int ch = 0; ch < nChunks; ++ch) {
    const int cur = ch & 1;
    if (ch + 1 < nChunks) {
      async_g2l_b128(p4 + (ch + 1) * 256 + tid, &stage[cur ^ 1][tid]);
      wait_asynccnt_le1();
    } else {
      wait_asynccnt_le0();
    }
    float4 v = stage[cur][tid];
    sum += expf(v.x - Mv) + expf(v.y - Mv) + expf(v.z - Mv) + expf(v.w - Mv);
  }
#else
  for (int i = tid; i < n4; i += 256) {
    if (((i >> 8) & 7) == 0 && i + 2048 < n4)
      __builtin_prefetch((const void*)(p4 + i + 2048), 0, 1); // global_prefetch_b8
    float4 v = p4[i];
    sum += expf(v.x - Mv) + expf(v.y - Mv) + expf(v.z - Mv) + expf(v.w - Mv);
  }
#endif
  red[tid] = sum;
  __syncthreads();
  for (int s = 128; s > 0; s >>= 1) {
    if (tid < s) red[tid] += red[tid + s];
    __syncthreads();
  }
  if (tid == 0) { Mout[m] = Mv; Zout[m] = red[0]; }
}

// =======================================================================
// Kernel 2: NMS + thresholds -> append candidates (score, flat idx).
// Monotone exp: 3x3 window-max test done on raw x. Thresholds:
//   xhat > 2/HW  and  exp(x - M) > 0.01 (since vmax = 1/Z).
// =======================================================================
__global__ void k_candidates(const float* __restrict__ pose,
                             const float* __restrict__ M,
                             const float* __restrict__ Z,
                             int* __restrict__ cnt,
                             float* __restrict__ cS,
                             int* __restrict__ cI) {
  const int m = blockIdx.y;
  const int p = blockIdx.x * blockDim.x + threadIdx.x;   // 0..HW-1
  const float* __restrict__ base = pose + (size_t)m * HWSZ;
  const int y = p >> 9, x = p & 511;
  const float v = base[p];

  bool ismax = true;
#pragma unroll
  for (int dy = -1; dy <= 1; ++dy) {
#pragma unroll
    for (int dx = -1; dx <= 1; ++dx) {
      if (dy == 0 && dx == 0) continue;
      int yy = y + dy, xx = x + dx;
      if (yy < 0 || yy >= HH || xx < 0 || xx >= WW) continue;
      if (base[yy * WW + xx] > v) ismax = false;
    }
  }
  if (!ismax) return;
  const float e = expf(v - M[m]);
  const float xhat = e / Z[m];
  if (!(xhat > THR_ABS) || !(e > THR_RELF)) return;

  int pos = atomicAdd(&cnt[m], 1);
  if (pos < CAP) {
    cS[m * CAP + pos] = xhat;
    cI[m * CAP + pos] = p;
  }
}

// =======================================================================
// Kernel 3: per-map top-K by (score desc, idx asc). One WG per map.
// Deterministic regardless of atomic append order.
// =======================================================================
__global__ void k_topk(const int* __restrict__ cnt,
                       const float* __restrict__ cS,
                       const int* __restrict__ cI,
                       float* __restrict__ out,
                       int* __restrict__ kx, int* __restrict__ ky,
                       int* __restrict__ kvalid) {
  const int m = blockIdx.x;
  const int tid = threadIdx.x;
  __shared__ float sS[CAP];
  __shared__ int   sI[CAP];
  __shared__ float rS[256];
  __shared__ int   rI[256];
  __shared__ int   rJ[256];

  int count = cnt[m];
  if (count > CAP) count = CAP;
  for (int j = tid; j < CAP; j += 256) {
    sS[j] = (j < count) ? cS[m * CAP + j] : -INFINITY;
    sI[j] = (j < count) ? cI[m * CAP + j] : 0x7fffffff;
  }
  __syncthreads();

  for (int k = 0; k < KK; ++k) {
    float bs = -INFINITY; int bi = 0x7fffffff; int bj = -1;
    for (int j = tid; j < CAP; j += 256) {
      float s = sS[j]; int ii = sI[j];
      if (s > bs || (s == bs && ii < bi)) { bs = s; bi = ii; bj = j; }
    }
    rS[tid] = bs; rI[tid] = bi; rJ[tid] = bj;
    __syncthreads();
    for (int t = 128; t > 0; t >>= 1) {
      if (tid < t) {
        float s = rS[tid + t]; int ii = rI[tid + t];
        if (s > rS[tid] || (s == rS[tid] && ii < rI[tid])) {
          rS[tid] = s; rI[tid] = ii; rJ[tid] = rJ[tid + t];
        }
      }
      __syncthreads();
    }
    if (tid == 0) {
      float s = rS[0]; int ii = rI[0]; int j = rJ[0];
      bool vld = (s > -INFINITY) && (j >= 0);
      int xx = vld ? (ii & 511) : 0;
      int yy = vld ? (ii >> 9) : 0;
      out[OFF_SCORES + (size_t)m * KK + k] = vld ? s : 0.f;
      out[OFF_COORDS + ((size_t)m * KK + k) * 2 + 0] = (float)xx;
      out[OFF_COORDS + ((size_t)m * KK + k) * 2 + 1] = (float)yy;
      out[OFF_VALID + (size_t)m * KK + k] = vld ? 1.f : 0.f;
      kx[m * KK + k] = xx; ky[m * KK + k] = yy; kvalid[m * KK + k] = vld ? 1 : 0;
      if (vld) sS[j] = -INFINITY;
    }
    __syncthreads();
  }
}

// =======================================================================
// Kernel 4: idx_pair (pure index function, data independent).
// =======================================================================
__global__ void k_idxpair(float* __restrict__ out) {
  int g = blockIdx.x * blockDim.x + threadIdx.x;
  if (g >= BB * AA * KK * KK) return;
  int c = g % KK;
  int r = (g / KK) % KK;
  int a = (g / (KK * KK)) % AA;
  int b = g / (KK * KK * AA);
  size_t base = OFF_IDXPAIR + (size_t)(b * AA + a) * 2 * (KK * KK);
  out[base + (size_t)r * KK + c]                   = (float)(a * KK + c);
  out[base + (size_t)KK * KK + (size_t)r * KK + c] = (float)((a + 1) * KK + r);
}

// =======================================================================
// Kernel 5: pairwise edges. One wave per 16x16 tile of the KxK pair grid.
// R^2 computed by V_WMMA_F32_16X16X4_F32:
//   A row  = [x2, y2, x2^2+y2^2, 1]       (rows  = i2 keypoints)
//   B col  = [-2x1, -2y1, 1, x1^2+y1^2]   (cols  = i1 keypoints)
//   dot    = (x2-x1)^2 + (y2-y1)^2        (exact: all integers < 2^24)
// Each lane then post-processes its 8 C/D elements (sqrt, validity,
// conditional 10-point PAF line integral).
// =======================================================================
__global__ void k_edges(const float* __restrict__ PAF,
                        const int* __restrict__ kx,
                        const int* __restrict__ ky,
                        const int* __restrict__ kvalid,
                        float* __restrict__ out) {
  const int a = blockIdx.y, b = blockIdx.z;
  const int wave = threadIdx.x >> 5;
  const int tile = blockIdx.x * 8 + wave;         // 7x7 = 49 tiles of 16x16
  if (tile >= 49) return;                          // wave-uniform branch
  const int lane = threadIdx.x & 31;
  const int tr = tile / 7, tc = tile % 7;
  const int m1 = b * SS + a;        // i1 map (columns)
  const int m2 = m1 + 1;            // i2 map (rows)
  const int l16 = lane & 15, kh = lane >> 4;

  // A fragment (p2 / rows)
  int r0 = tr * 16 + l16;
  float x2 = 0.f, y2 = 0.f;
  if (r0 < KK) { x2 = (float)kx[m2 * KK + r0]; y2 = (float)ky[m2 * KK + r0]; }
  v2f af;
  if (kh == 0) { af.x = x2; af.y = y2; }
  else         { af.x = x2 * x2 + y2 * y2; af.y = 1.f; }

  // B fragment (p1 / cols)
  int c0 = tc * 16 + l16;
  float x1 = 0.f, y1 = 0.f;
  if (c0 < KK) { x1 = (float)kx[m1 * KK + c0]; y1 = (float)ky[m1 * KK + c0]; }
  v2f bf;
  if (kh == 0) { bf.x = -2.f * x1; bf.y = -2.f * y1; }
  else         { bf.x = 1.f; bf.y = x1 * x1 + y1 * y1; }

  v8f c8 = {0.f, 0.f, 0.f, 0.f, 0.f, 0.f, 0.f, 0.f};
  v8f d;
#if defined(__HIP_DEVICE_COMPILE__)
  d = __builtin_amdgcn_wmma_f32_16x16x4_f32(false, af, false, bf,
                                            (short)0, c8, false, false);
#else
  d = c8;
#endif

  const float* __restrict__ pafx = PAF + ((size_t)(b * AA + a) * 2) * HWSZ;
  const float* __restrict__ pafy = pafx + (size_t)HWSZ;

#pragma unroll
  for (int j = 0; j < 8; ++j) {
    int r = tr * 16 + (j + 8 * kh);   // C/D layout: vgpr j, lanes16-31 -> M=8+j
    int c = tc * 16 + l16;
    if (r >= KK || c >= KK) continue;
    int X1 = kx[m1 * KK + c], Y1 = ky[m1 * KK + c];
    int X2 = kx[m2 * KK + r], Y2 = ky[m2 * KK + r];
    int v1 = kvalid[m1 * KK + c], v2k = kvalid[m2 * KK + r];

    float R = sqrtf(fmaxf(d[j], 0.f));
    bool pv = (R < KP_MAX_DISTF) && (v1 != 0) && (v2k != 0);
    float li = 0.f, eR = 0.f;
    if (pv) {
      eR = R;
      if (R > 0.f) {
        float tvx = (float)(X2 - X1) / R;
        float tvy = (float)(Y2 - Y1) / R;
        float acc = 0.f;
#pragma unroll
        for (int t = 0; t < NPTS; ++t) {
          int ix = (X1 * (NPTS - 1 - t) + X2 * t) / (NPTS - 1);
          int iy = (Y1 * (NPTS - 1 - t) + Y2 * t) / (NPTS - 1);
          acc += tvx * pafx[iy * WW + ix] + tvy * pafy[iy * WW + ix];
        }
        li = acc * (1.f / NPTS);
      }
    }
    size_t eo = ((size_t)(b * AA + a) * KK + r) * KK + c;
    out[OFF_EPAF + eo] = li;
    out[OFF_ER + eo] = eR;
    out[OFF_PV + eo] = pv ? 1.f : 0.f;
  }
}

// =======================================================================
extern "C" void kernel_launch(void* const* d_in, const int* in_sizes, int n_in,
                              void* d_out, int out_size, void* d_ws, size_t ws_size,
                              hipStream_t stream) {
  const float* pose = (const float*)d_in[0];
  const float* paf  = (const float*)d_in[1];
  float* out = (float*)d_out;

  // workspace carve-out (~1.9 MB)
  char* ws = (char*)d_ws;
  float* M    = (float*)ws;                    // NMAPS
  float* Z    = M + NMAPS;                     // NMAPS
  int*   cnt  = (int*)(Z + NMAPS);             // NMAPS
  float* cS   = (float*)(cnt + NMAPS);         // NMAPS*CAP
  int*   cI   = (int*)(cS + (size_t)NMAPS * CAP);
  int*   kxp  = cI + (size_t)NMAPS * CAP;      // NMAPS*K
  int*   kyp  = kxp + NMAPS * KK;
  int*   kvp  = kyp + NMAPS * KK;

  (void)hipMemsetAsync(cnt, 0, NMAPS * sizeof(int), stream);

  k_softmax_stats<<<NMAPS, 256, 0, stream>>>(pose, M, Z);
  k_candidates<<<dim3(HWSZ / 256, NMAPS), 256, 0, stream>>>(pose, M, Z, cnt, cS, cI);
  k_topk<<<NMAPS, 256, 0, stream>>>(cnt, cS, cI, out, kxp, kyp, kvp);
  k_idxpair<<<(BB * AA * KK * KK + 255) / 256, 256, 0, stream>>>(out);
  k_edges<<<dim3(7, AA, BB), 256, 0, stream>>>(paf, kxp, kyp, kvp, out);
}